// myloss_26482768347185
// MI455X (gfx1250) — compile-verified
//
#include <hip/hip_runtime.h>
#include <math.h>

typedef float v2f __attribute__((ext_vector_type(2)));
typedef float v4f __attribute__((ext_vector_type(4)));
typedef float v8f __attribute__((ext_vector_type(8)));

#define FEATN 4096
#define ROWSN 4096
#define WAVES_PER_BLOCK 8
#define BLOCK_THREADS (WAVES_PER_BLOCK * 32)

// Kernel 1: one wave32 per row. Streams out1/out2/out3 rows with coalesced
// non-temporal b128 loads, accumulates per-lane partial squared distances,
// then reduces across the wave with V_WMMA_F32_16X16X4_F32 (exact f32 math:
// A = 16x4 of lane partials, B = 0/1 selector => D[m,n] = s[m] + s[m+16]).
__global__ __launch_bounds__(BLOCK_THREADS)
void row_hinge_kernel(const float* __restrict__ out1,
                      const float* __restrict__ out2,
                      const float* __restrict__ out3,
                      float* __restrict__ row_relu)
{
    const int lane = threadIdx.x & 31;
    const int wave = threadIdx.x >> 5;
    const int row  = blockIdx.x * WAVES_PER_BLOCK + wave;
    const size_t base = (size_t)row * FEATN;

    const v4f* __restrict__ p1 = (const v4f*)(out1 + base);
    const v4f* __restrict__ p2 = (const v4f*)(out2 + base);
    const v4f* __restrict__ p3 = (const v4f*)(out3 + base);

    v4f s12 = {0.f, 0.f, 0.f, 0.f};
    v4f s13 = {0.f, 0.f, 0.f, 0.f};

    // 4096 floats / row = 1024 v4f; 32 lanes -> 32 iterations, 512B/array/iter.
#pragma unroll 4
    for (int it = 0; it < 32; ++it) {
        const int idx = it * 32 + lane;
        // Prefetch 8 iterations (4 KB) ahead; speculative, OOB is dropped.
        __builtin_prefetch((const void*)(p1 + idx + 256), 0, 0);
        __builtin_prefetch((const void*)(p2 + idx + 256), 0, 0);
        __builtin_prefetch((const void*)(p3 + idx + 256), 0, 0);
        v4f a = __builtin_nontemporal_load(p1 + idx);
        v4f b = __builtin_nontemporal_load(p2 + idx);
        v4f c = __builtin_nontemporal_load(p3 + idx);
        v4f d12 = a - b;
        v4f d13 = a - c;
        s12 += d12 * d12;   // packed f32 FMAs
        s13 += d13 * d13;
    }

    float s12s = (s12[0] + s12[1]) + (s12[2] + s12[3]);
    float s13s = (s13[0] + s13[1]) + (s13[2] + s13[3]);

    // ---- wave-wide reduction via f32 WMMA ----
    // A-layout (16x4 f32): lanes 0-15 hold K0/K1 in vgpr0/vgpr1 (M = lane),
    // lanes 16-31 hold K2/K3 (M = lane-16). Pack s12 -> K even, s13 -> K odd.
    v2f aop; aop[0] = s12s; aop[1] = s13s;
    v2f bsel12; bsel12[0] = 1.0f; bsel12[1] = 0.0f;  // select K0,K2 rows
    v2f bsel13; bsel13[0] = 0.0f; bsel13[1] = 1.0f;  // select K1,K3 rows
    v8f zero = {};

    v8f m12 = __builtin_amdgcn_wmma_f32_16x16x4_f32(
        false, aop, false, bsel12, (short)0, zero, false, false);
    v8f m13 = __builtin_amdgcn_wmma_f32_16x16x4_f32(
        false, aop, false, bsel13, (short)0, zero, false, false);

    // D vgpr r: lanes 0-15 hold M=r, lanes 16-31 hold M=r+8 (all columns equal).
    float y12 = ((m12[0] + m12[1]) + (m12[2] + m12[3])) +
                ((m12[4] + m12[5]) + (m12[6] + m12[7]));
    float y13 = ((m13[0] + m13[1]) + (m13[2] + m13[3])) +
                ((m13[4] + m13[5]) + (m13[6] + m13[7]));
    // Combine the two lane-halves (M=0..7 sums live in lanes 0-15, M=8..15 in 16-31).
    y12 += __shfl_xor(y12, 16, 32);
    y13 += __shfl_xor(y13, 16, 32);

    if (lane == 0) {
        float compare = 2.0f - sqrtf(y13) + sqrtf(y12);
        row_relu[row] = fmaxf(compare, 0.0f);
    }
}

// Kernel 2: deterministic fixed-order reduction of the 4096 per-row hinge
// values; output scalar = B * sum (the [B,B] broadcast in the reference).
__global__ __launch_bounds__(256)
void final_reduce_kernel(const float* __restrict__ row_relu,
                         float* __restrict__ out)
{
    __shared__ float sm[256];
    float s = 0.0f;
    for (int i = threadIdx.x; i < ROWSN; i += 256)
        s += row_relu[i];
    sm[threadIdx.x] = s;
    __syncthreads();
    for (int off = 128; off > 0; off >>= 1) {
        if (threadIdx.x < off)
            sm[threadIdx.x] += sm[threadIdx.x + off];
        __syncthreads();
    }
    if (threadIdx.x == 0)
        out[0] = 4096.0f * sm[0];
}

extern "C" void kernel_launch(void* const* d_in, const int* in_sizes, int n_in,
                              void* d_out, int out_size, void* d_ws, size_t ws_size,
                              hipStream_t stream)
{
    (void)in_sizes; (void)n_in; (void)out_size; (void)ws_size;
    const float* out1 = (const float*)d_in[0];
    const float* out2 = (const float*)d_in[1];
    const float* out3 = (const float*)d_in[2];
    float* ws  = (float*)d_ws;   // 4096 floats of scratch
    float* out = (float*)d_out;

    row_hinge_kernel<<<ROWSN / WAVES_PER_BLOCK, BLOCK_THREADS, 0, stream>>>(
        out1, out2, out3, ws);
    final_reduce_kernel<<<1, 256, 0, stream>>>(ws, out);
}